// VGAE_64063732187138
// MI455X (gfx1250) — compile-verified
//
#include <hip/hip_runtime.h>
#include <hip/hip_bf16.h>

#define N_NODES 100000
#define N_EDGES 1600000
#define IN_F    256
#define H0      128
#define H1      64
#define SCAN_B  1024
#define N_SCANBLK ((N_NODES + SCAN_B - 1) / SCAN_B)   // 98

typedef __attribute__((ext_vector_type(2))) float v2f;
typedef __attribute__((ext_vector_type(8))) float v8f;

// ---------------- utility ----------------

__global__ void k_zero_i(int* __restrict__ p, int n) {
  int i = blockIdx.x * blockDim.x + threadIdx.x;
  int stride = gridDim.x * blockDim.x;
  for (; i < n; i += stride) p[i] = 0;
}

// int degree histograms: degs[0..N) = out-degree(src), degs[N..2N) = in-degree(dst)
__global__ void k_deg(const int* __restrict__ src, const int* __restrict__ dst,
                      int* __restrict__ degs) {
  int e = blockIdx.x * blockDim.x + threadIdx.x;
  if (e < N_EDGES) {
    atomicAdd(&degs[src[e]], 1);               // global_atomic_add_u32
    atomicAdd(&degs[N_NODES + dst[e]], 1);
  }
}

// norms[i] = rsqrt(max(deg,1)) for both halves (2N elements)
__global__ void k_norm(const int* __restrict__ degs, float* __restrict__ norms, int n) {
  int i = blockIdx.x * blockDim.x + threadIdx.x;
  if (i < n) norms[i] = rsqrtf(fmaxf((float)degs[i], 1.0f));
}

// ---------------- 3-kernel exclusive scan of in-degrees -> row_ptr ----------------

__global__ __launch_bounds__(SCAN_B) void k_scan_blk(const int* __restrict__ deg,
                                                     int* __restrict__ excl,
                                                     int* __restrict__ blksum) {
  __shared__ int buf[SCAN_B];
  const int tid = threadIdx.x;
  const int i = blockIdx.x * SCAN_B + tid;
  const int v = (i < N_NODES) ? deg[i] : 0;
  buf[tid] = v;
  __syncthreads();
  for (int off = 1; off < SCAN_B; off <<= 1) {
    int t = (tid >= off) ? buf[tid - off] : 0;
    __syncthreads();
    buf[tid] += t;
    __syncthreads();
  }
  if (i < N_NODES) excl[i] = buf[tid] - v;     // exclusive within block
  if (tid == 0) blksum[blockIdx.x] = buf[SCAN_B - 1];
}

__global__ void k_scan_top(const int* __restrict__ blksum, int* __restrict__ blkoff) {
  if (blockIdx.x == 0 && threadIdx.x == 0) {
    int run = 0;
    for (int b = 0; b < N_SCANBLK; ++b) { blkoff[b] = run; run += blksum[b]; }
  }
}

__global__ void k_scan_add(int* __restrict__ row_ptr, const int* __restrict__ blkoff,
                           int* __restrict__ cursor) {
  int i = blockIdx.x * blockDim.x + threadIdx.x;
  if (i < N_NODES) {
    int rp = row_ptr[i] + blkoff[i >> 10];
    row_ptr[i] = rp;
    cursor[i] = rp;
  }
  if (i == 0) row_ptr[N_NODES] = N_EDGES;
}

// bucket edges by destination: csr_src[row_ptr[d] ...] = sources of d's in-edges
__global__ void k_fill(const int* __restrict__ src, const int* __restrict__ dst,
                       int* __restrict__ cursor, int* __restrict__ csr) {
  int e = blockIdx.x * blockDim.x + threadIdx.x;
  if (e < N_EDGES) {
    int pos = atomicAdd(&cursor[dst[e]], 1);
    csr[pos] = src[e];
  }
}

// ---------------- WMMA GEMM 0: hpre = (x * onorm) @ W0  [N,256]x[256,128] ----------------
// 256 threads = 8 waves; block computes a 16(M) x 128(N) strip, one 16x16 tile per wave.
// A staged in LDS, stride 260 (260 mod 64 == 4 -> conflict-free column reads).

__global__ __launch_bounds__(256) void k_gemm0(const float* __restrict__ x,
                                               const float* __restrict__ W0,
                                               const float* __restrict__ onorm,
                                               float* __restrict__ hpre) {
  __shared__ float As[16 * 260];
  const int tid = threadIdx.x;
  const int m0 = blockIdx.x * 16;

  for (int i = tid; i < 16 * 64; i += 256) {
    const int row = i >> 6;
    const int c4 = (i & 63) << 2;
    const float s = onorm[m0 + row];
    const float4 v = *(const float4*)(x + (size_t)(m0 + row) * IN_F + c4);
    *(float4*)(&As[row * 260 + c4]) = make_float4(v.x * s, v.y * s, v.z * s, v.w * s);
  }
  __syncthreads();

  const int lane = tid & 31;
  const int wave = tid >> 5;
  const int col = wave * 16 + (lane & 15);
  const int arow = lane & 15;
  const int khalf = (lane >> 4) << 1;          // lanes 0-15: K+0,1 ; lanes 16-31: K+2,3

  v8f acc = {};
  for (int k = 0; k < IN_F; k += 4) {
    const int kk = k + khalf;
    v2f a, b;
    a.x = As[arow * 260 + kk];
    a.y = As[arow * 260 + kk + 1];
    b.x = W0[kk * H0 + col];
    b.y = W0[(kk + 1) * H0 + col];
    acc = __builtin_amdgcn_wmma_f32_16x16x4_f32(false, a, false, b, (short)0, acc,
                                                false, false);
  }

  const int rbase = (lane >> 4) << 3;
  for (int r = 0; r < 8; ++r)
    hpre[(size_t)(m0 + rbase + r) * H0 + col] = acc[r];
}

// ------- CSR gather-reduce, layer 0 (no atomics) + fused relu(agg*in_norm+b0) -------
// One wave per node; lane = 4-col quad (32 quads = 128 cols). Per edge: one broadcast
// index load + a fully coalesced 512B row read (L2-resident).

__global__ __launch_bounds__(256) void k_agg0(const int* __restrict__ row_ptr,
                                              const int* __restrict__ csr,
                                              const float* __restrict__ hpre,
                                              const float* __restrict__ inorm,
                                              const float* __restrict__ b0,
                                              float* __restrict__ h) {
  const int tid = threadIdx.x;
  const int node = blockIdx.x * 8 + (tid >> 5);
  const int c = (tid & 31) << 2;
  const int beg = row_ptr[node];
  const int end = row_ptr[node + 1];
  float ax = 0.f, ay = 0.f, az = 0.f, aw = 0.f;
  for (int k = beg; k < end; ++k) {
    const int s = csr[k];
    const float4 v = *(const float4*)(hpre + (size_t)s * H0 + c);
    ax += v.x; ay += v.y; az += v.z; aw += v.w;
  }
  const float sc = inorm[node];
  const float4 bb = *(const float4*)(b0 + c);
  float4 r;
  r.x = fmaxf(ax * sc + bb.x, 0.0f);
  r.y = fmaxf(ay * sc + bb.y, 0.0f);
  r.z = fmaxf(az * sc + bb.z, 0.0f);
  r.w = fmaxf(aw * sc + bb.w, 0.0f);
  *(float4*)(h + (size_t)node * H0 + c) = r;
}

// ---------------- WMMA GEMMs 1&2 fused: mpre=(h*onorm)@W1, spre=(h*onorm)@W2 ----------------

__global__ __launch_bounds__(256) void k_gemm12(const float* __restrict__ h,
                                                const float* __restrict__ W1,
                                                const float* __restrict__ W2,
                                                const float* __restrict__ onorm,
                                                float* __restrict__ mpre,
                                                float* __restrict__ spre) {
  __shared__ float As[16 * 132];
  const int tid = threadIdx.x;
  const int m0 = blockIdx.x * 16;

  for (int i = tid; i < 16 * 32; i += 256) {
    const int row = i >> 5;
    const int c4 = (i & 31) << 2;
    const float s = onorm[m0 + row];
    const float4 v = *(const float4*)(h + (size_t)(m0 + row) * H0 + c4);
    *(float4*)(&As[row * 132 + c4]) = make_float4(v.x * s, v.y * s, v.z * s, v.w * s);
  }
  __syncthreads();

  const int lane = tid & 31;
  const int wave = tid >> 5;
  const float* W = (wave < 4) ? W1 : W2;       // wave-uniform select, no divergence
  float* outp = (wave < 4) ? mpre : spre;
  const int col = (wave & 3) * 16 + (lane & 15);
  const int arow = lane & 15;
  const int khalf = (lane >> 4) << 1;

  v8f acc = {};
  for (int k = 0; k < H0; k += 4) {
    const int kk = k + khalf;
    v2f a, b;
    a.x = As[arow * 132 + kk];
    a.y = As[arow * 132 + kk + 1];
    b.x = W[kk * H1 + col];
    b.y = W[(kk + 1) * H1 + col];
    acc = __builtin_amdgcn_wmma_f32_16x16x4_f32(false, a, false, b, (short)0, acc,
                                                false, false);
  }

  const int rbase = (lane >> 4) << 3;
  for (int r = 0; r < 8; ++r)
    outp[(size_t)(m0 + rbase + r) * H1 + col] = acc[r];
}

// ------- CSR gather-reduce for mean & log_std + fused out = mean + noise*exp(log_std) -------
// 16 lanes per node (16 quads = 64 cols); each lane accumulates BOTH mean and log_std quads.

__global__ __launch_bounds__(256) void k_agg12(const int* __restrict__ row_ptr,
                                               const int* __restrict__ csr,
                                               const float* __restrict__ mpre,
                                               const float* __restrict__ spre,
                                               const float* __restrict__ inorm,
                                               const float* __restrict__ b1,
                                               const float* __restrict__ b2,
                                               const float* __restrict__ noise,
                                               float* __restrict__ out) {
  const int tid = threadIdx.x;
  const int node = blockIdx.x * 16 + (tid >> 4);
  const int c = (tid & 15) << 2;
  const int beg = row_ptr[node];
  const int end = row_ptr[node + 1];
  float mx = 0.f, my = 0.f, mz = 0.f, mw = 0.f;
  float sx = 0.f, sy = 0.f, sz = 0.f, sw = 0.f;
  for (int k = beg; k < end; ++k) {
    const int s = csr[k];
    const float4 vm = *(const float4*)(mpre + (size_t)s * H1 + c);
    const float4 vs = *(const float4*)(spre + (size_t)s * H1 + c);
    mx += vm.x; my += vm.y; mz += vm.z; mw += vm.w;
    sx += vs.x; sy += vs.y; sz += vs.z; sw += vs.w;
  }
  const float sc = inorm[node];
  const float4 B1 = *(const float4*)(b1 + c);
  const float4 B2 = *(const float4*)(b2 + c);
  const float4 nz = *(const float4*)(noise + (size_t)node * H1 + c);
  float4 r;
  r.x = (mx * sc + B1.x) + nz.x * expf(sx * sc + B2.x);
  r.y = (my * sc + B1.y) + nz.y * expf(sy * sc + B2.y);
  r.z = (mz * sc + B1.z) + nz.z * expf(sz * sc + B2.z);
  r.w = (mw * sc + B1.w) + nz.w * expf(sw * sc + B2.w);
  *(float4*)(out + (size_t)node * H1 + c) = r;
}

// ---------------- launch ----------------

extern "C" void kernel_launch(void* const* d_in, const int* in_sizes, int n_in,
                              void* d_out, int out_size, void* d_ws, size_t ws_size,
                              hipStream_t stream) {
  const float* x     = (const float*)d_in[0];
  const int*   src   = (const int*)d_in[1];
  const int*   dst   = (const int*)d_in[2];
  const float* noise = (const float*)d_in[3];
  const float* W0    = (const float*)d_in[4];
  const float* b0    = (const float*)d_in[5];
  const float* W1    = (const float*)d_in[6];
  const float* b1    = (const float*)d_in[7];
  const float* W2    = (const float*)d_in[8];
  const float* b2    = (const float*)d_in[9];
  float* out = (float*)d_out;

  // workspace layout
  char* wsb = (char*)d_ws;
  int*   degs    = (int*)wsb;                           wsb += sizeof(int) * 2 * N_NODES;
  float* norms   = (float*)wsb;                         wsb += sizeof(float) * 2 * N_NODES;
  int*   row_ptr = (int*)wsb;                           wsb += sizeof(int) * (N_NODES + 1);
  int*   cursor  = (int*)wsb;                           wsb += sizeof(int) * N_NODES;
  int*   blksum  = (int*)wsb;                           wsb += sizeof(int) * 128;
  int*   blkoff  = (int*)wsb;                           wsb += sizeof(int) * 128;
  int*   csr     = (int*)wsb;                           wsb += sizeof(int) * N_EDGES;
  wsb = (char*)(((size_t)wsb + 255) & ~(size_t)255);
  float* buf1    = (float*)wsb;                         wsb += sizeof(float) * (size_t)N_NODES * H0;
  float* buf2    = (float*)wsb;                         // N*H0 floats
  float* onorm = norms;
  float* inorm = norms + N_NODES;
  float* mpre  = buf1;
  float* spre  = buf1 + (size_t)N_NODES * H1;

  // 1. degree histograms (int atomics)
  k_zero_i<<<1024, 256, 0, stream>>>(degs, 2 * N_NODES);
  k_deg<<<N_EDGES / 256, 256, 0, stream>>>(src, dst, degs);
  // 2. norms = rsqrt(max(deg,1))
  k_norm<<<(2 * N_NODES + 255) / 256, 256, 0, stream>>>(degs, norms, 2 * N_NODES);
  // 3. exclusive scan of in-degrees -> row_ptr, cursor
  k_scan_blk<<<N_SCANBLK, SCAN_B, 0, stream>>>(degs + N_NODES, row_ptr, blksum);
  k_scan_top<<<1, 32, 0, stream>>>(blksum, blkoff);
  k_scan_add<<<(N_NODES + 255) / 256, 256, 0, stream>>>(row_ptr, blkoff, cursor);
  // 4. bucket edges by dst
  k_fill<<<N_EDGES / 256, 256, 0, stream>>>(src, dst, cursor, csr);
  // 5. GEMM0 (WMMA f32): buf1 = (x*onorm) @ W0
  k_gemm0<<<N_NODES / 16, 256, 0, stream>>>(x, W0, onorm, buf1);
  // 6. gather-reduce layer 0 + relu epilogue -> buf2 = h
  k_agg0<<<N_NODES / 8, 256, 0, stream>>>(row_ptr, csr, buf1, inorm, b0, buf2);
  // 7. GEMMs 1&2 (WMMA f32): buf1 halves = (h*onorm) @ {W1,W2}
  k_gemm12<<<N_NODES / 16, 256, 0, stream>>>(buf2, W1, W2, onorm, mpre, spre);
  // 8. gather-reduce both + fused sampling epilogue -> out
  k_agg12<<<N_NODES / 16, 256, 0, stream>>>(row_ptr, csr, mpre, spre, inorm, b1, b2,
                                            noise, out);
}